// SimpleFourierLoss_32220844655149
// MI455X (gfx1250) — compile-verified
//
#include <hip/hip_runtime.h>
#include <math.h>
#include <stdint.h>

// ---------------- problem constants ----------------
#define BATCH  8
#define SL     16                 // 16 slices: 0..7 = pred, 8..15 = target
#define HH     512
#define WW     512
#define PH     514
#define PW     514
#define SLC    (PH*PW)            // 264196 padded pixels per slice
#define IMG    (HH*WW)            // 262144 interior pixels
#define TS     16384              // hash table slots per slice (power of 2)
#define ATS    (TS+1)             // +1 slot for background label 0
#define MAXC   256                // max kept components per slice (multiple of 16)
#define MINPTS 10
#define SUMT   66977792.0         // sum over all interior pixels of row idx (== col idx sum)

// temporal blocking: 128x128 LDS tile, halo 16 -> 16 local iterations/launch
#define TILE   128
#define HALO   16
#define CORE   96
#define NTIL   6                  // 6*96 = 576 >= 512

typedef float v2f __attribute__((ext_vector_type(2)));
typedef float v8f __attribute__((ext_vector_type(8)));

// ---- CDNA5 async global->LDS path (guarded; falls back to plain loads) ----
#if defined(__has_builtin)
#  if __has_builtin(__builtin_amdgcn_global_load_async_to_lds_b32) && \
      __has_builtin(__builtin_amdgcn_s_wait_asynccnt)
#    define USE_ASYNC_LDS 1
#  endif
#endif
#ifndef USE_ASYNC_LDS
#define USE_ASYNC_LDS 0
#endif

#if USE_ASYNC_LDS
typedef __attribute__((address_space(1))) int gas_int;   // global
typedef __attribute__((address_space(3))) int las_int;   // LDS
#endif

// ============================================================
// K1: seed padded label images for pred and target.
//     mask==1 cells get the reference's exact linear index
//     idx = (b*514 + y)*514 + x  (exact in f32, max < 2^24).
// ============================================================
__global__ void k_init(const float* __restrict__ pred, const float* __restrict__ tgt,
                       float* __restrict__ w0, float* __restrict__ w1) {
    long long i = (long long)blockIdx.x * blockDim.x + threadIdx.x;
    if (i >= (long long)SL * SLC) return;
    int s = (int)(i / SLC);
    int r = (int)(i % SLC);
    int y = r / PW, x = r % PW;
    float v = 0.f;
    if (y >= 1 && y <= HH && x >= 1 && x <= WW) {
        int b = s & 7;
        bool m;
        size_t off = (size_t)b * IMG + (size_t)(y - 1) * WW + (x - 1);
        if (s < 8) m = (pred[off] > 0.0f);            // sigmoid(x)>0.5  <=>  x>0
        else       m = (tgt[off] == 1.0f);            // mask = (I == 1.0)
        if (m) v = (float)((b * PH + y) * PW + x);
    }
    w0[i] = v;
    w1[i] = v;
}

// ============================================================
// K2: temporally-blocked masked 3x3 max propagation.
//     out = (center>0) ? max3x3(in) : 0   (mask == value>0 invariant)
//     128x128 LDS tile, separable row/col max, `iters` <= HALO local steps,
//     core 96x96 written back. OOB-of-padded-array reads are 0 (exact).
// ============================================================
__global__ __launch_bounds__(256) void k_prop(const float* __restrict__ src,
                                              float* __restrict__ dst, int iters) {
    __shared__ float A[TILE * TILE];
    __shared__ float Bv[TILE * TILE];
    const int s  = blockIdx.z;
    const int oy = 1 + blockIdx.y * CORE - HALO;   // tile load origin (may be <0)
    const int ox = 1 + blockIdx.x * CORE - HALO;
    const int tx = threadIdx.x, ty = threadIdx.y;
    const float* sp = src + (size_t)s * SLC;

    // ---- fill tile (async global->LDS when available) ----
    #pragma unroll
    for (int cy = 0; cy < 8; ++cy) {
        #pragma unroll
        for (int cx = 0; cx < 8; ++cx) {
            int ly = ty + cy * 16, lx = tx + cx * 16;
            int gy = oy + ly, gx = ox + lx;
            bool inb = (gy >= 0) && (gy < PH) && (gx >= 0) && (gx < PW);
#if USE_ASYNC_LDS
            if (inb) {
                __builtin_amdgcn_global_load_async_to_lds_b32(
                    (gas_int*)(sp + (size_t)gy * PW + gx),
                    (las_int*)&A[ly * TILE + lx], 0, 0);
            } else {
                A[ly * TILE + lx] = 0.f;
            }
#else
            A[ly * TILE + lx] = inb ? sp[(size_t)gy * PW + gx] : 0.f;
#endif
        }
    }
#if USE_ASYNC_LDS
    __builtin_amdgcn_s_wait_asynccnt(0);
#endif

    for (int it = 0; it < iters; ++it) {
        __syncthreads();
        // pass 1: horizontal max (includes center) -> Bv
        #pragma unroll
        for (int cy = 0; cy < 8; ++cy)
            #pragma unroll
            for (int cx = 0; cx < 8; ++cx) {
                int ly = ty + cy * 16, lx = tx + cx * 16;
                float c = A[ly * TILE + lx];
                float l = (lx > 0)        ? A[ly * TILE + lx - 1] : 0.f;
                float r = (lx < TILE - 1) ? A[ly * TILE + lx + 1] : 0.f;
                Bv[ly * TILE + lx] = fmaxf(c, fmaxf(l, r));
            }
        __syncthreads();
        // pass 2: vertical max + mask, in place into A (each thread touches only its own center)
        #pragma unroll
        for (int cy = 0; cy < 8; ++cy)
            #pragma unroll
            for (int cx = 0; cx < 8; ++cx) {
                int ly = ty + cy * 16, lx = tx + cx * 16;
                float c = A[ly * TILE + lx];
                float m = Bv[ly * TILE + lx];
                float u = (ly > 0)        ? Bv[(ly - 1) * TILE + lx] : 0.f;
                float d = (ly < TILE - 1) ? Bv[(ly + 1) * TILE + lx] : 0.f;
                A[ly * TILE + lx] = (c > 0.f) ? fmaxf(m, fmaxf(u, d)) : 0.f;
            }
    }
    __syncthreads();

    // ---- write back core (exact region after `iters` <= HALO steps) ----
    float* dp = dst + (size_t)s * SLC;
    #pragma unroll
    for (int cy = 0; cy < 6; ++cy)
        #pragma unroll
        for (int cx = 0; cx < 6; ++cx) {
            int ly = HALO + ty + cy * 16, lx = HALO + tx + cx * 16;
            int gy = oy + ly, gx = ox + lx;
            if (gy >= 1 && gy < PH - 1 && gx >= 1 && gx < PW - 1)
                dp[(size_t)gy * PW + gx] = A[ly * TILE + lx];
        }
}

// ---------------- per-slice open-addressing hash ----------------
__device__ __forceinline__ int hash_slot(unsigned key) {
    return (int)((key * 2654435761u) & (TS - 1));
}
__device__ int hash_insert(unsigned* keys, unsigned key) {
    int h = hash_slot(key);
    for (int p = 0; p < TS; ++p) {
        unsigned k = keys[h];
        if (k == key) return h;
        if (k == 0u) {
            unsigned old = atomicCAS(&keys[h], 0u, key);
            if (old == 0u || old == key) return h;
        }
        h = (h + 1) & (TS - 1);
    }
    return -1;
}
__device__ int hash_find(const unsigned* keys, unsigned key) {
    int h = hash_slot(key);
    for (int p = 0; p < TS; ++p) {
        unsigned k = keys[h];
        if (k == key) return h;
        if (k == 0u) return -1;
        h = (h + 1) & (TS - 1);
    }
    return -1;
}

// ============================================================
// K3: boundary test + contour labels + per-label count/sum accumulation.
//     boundary <=> center masked (cc>0) and some 8-neighbor unmasked (cc==0).
// ============================================================
__global__ void k_contour(const float* __restrict__ cc, float* __restrict__ lab,
                          unsigned* hkey, unsigned* hcnt, unsigned* hsr, unsigned* hsc,
                          unsigned* nb, unsigned* bsr, unsigned* bsc) {
    long long i = (long long)blockIdx.x * blockDim.x + threadIdx.x;
    if (i >= (long long)SL * IMG) return;
    int s  = (int)(i / IMG);
    int r  = (int)(i % IMG);
    int ry = r / WW, rx = r % WW;
    int y = ry + 1, x = rx + 1;
    const float* p = cc + (size_t)s * SLC;
    float v = p[(size_t)y * PW + x];
    float out = 0.f;
    if (v > 0.f) {
        bool bnd = false;
        #pragma unroll
        for (int dy = -1; dy <= 1; ++dy)
            #pragma unroll
            for (int dx = -1; dx <= 1; ++dx) {
                if (dy == 0 && dx == 0) continue;
                if (p[(size_t)(y + dy) * PW + (x + dx)] == 0.f) bnd = true;
            }
        if (bnd) out = v;
    }
    lab[i] = out;
    if (out > 0.f) {
        int e = hash_insert(hkey + (size_t)s * TS, __float_as_uint(out));
        if (e >= 0) {
            atomicAdd(hcnt + (size_t)s * TS + e, 1u);
            atomicAdd(hsr  + (size_t)s * TS + e, (unsigned)ry);
            atomicAdd(hsc  + (size_t)s * TS + e, (unsigned)rx);
        }
        atomicAdd(nb + s, 1u);
        atomicAdd(bsr + s, (unsigned)ry);
        atomicAdd(bsc + s, (unsigned)rx);
    }
}

// monotone double -> uint64 encoding for atomicMin on angle
__device__ __forceinline__ unsigned long long enc_d(double a) {
    long long b = __double_as_longlong(a);
    return (b < 0) ? ~(unsigned long long)b
                   : ((unsigned long long)b | 0x8000000000000000ULL);
}

// resolve (entry index, centroid) for a pixel's label; entry TS == background
__device__ bool entry_centroid(float lbl, int s,
                               const unsigned* hkey, const unsigned* hcnt,
                               const unsigned* hsr, const unsigned* hsc,
                               const unsigned* nb, const unsigned* bsr, const unsigned* bsc,
                               int* e_out, double* crow, double* ccol) {
    if (lbl == 0.f) {
        unsigned n = nb[s];
        if (n == 0u) return false;                  // slice has no contours -> skipped
        unsigned c = (unsigned)IMG - n;
        *e_out = TS;
        *crow = (SUMT - (double)bsr[s]) / (double)c;
        *ccol = (SUMT - (double)bsc[s]) / (double)c;
        return true;
    }
    int e = hash_find(hkey + (size_t)s * TS, __float_as_uint(lbl));
    if (e < 0) return false;
    unsigned c = hcnt[(size_t)s * TS + e];
    if (c == 0u) return false;
    *e_out = e;
    *crow = (double)hsr[(size_t)s * TS + e] / (double)c;
    *ccol = (double)hsc[(size_t)s * TS + e] / (double)c;
    return true;
}

// K4a: per-label minimum of float64 atan2(col-cc, row-cr)
__global__ void k_angmin(const float* __restrict__ lab,
                         const unsigned* hkey, const unsigned* hcnt,
                         const unsigned* hsr, const unsigned* hsc,
                         const unsigned* nb, const unsigned* bsr, const unsigned* bsc,
                         unsigned long long* amin) {
    long long i = (long long)blockIdx.x * blockDim.x + threadIdx.x;
    if (i >= (long long)SL * IMG) return;
    int s = (int)(i / IMG);
    int r = (int)(i % IMG);
    int ry = r / WW, rx = r % WW;
    float lbl = lab[i];
    int e; double cr, cc2;
    if (!entry_centroid(lbl, s, hkey, hcnt, hsr, hsc, nb, bsr, bsc, &e, &cr, &cc2)) return;
    double ang = atan2((double)rx - cc2, (double)ry - cr);
    atomicMin(&amin[(size_t)s * ATS + e], enc_d(ang));
}

// K4b: among pixels achieving the minimum angle, take smallest row-major index
//      (== stable argsort tie-break over np.argwhere order)
__global__ void k_idxmin(const float* __restrict__ lab,
                         const unsigned* hkey, const unsigned* hcnt,
                         const unsigned* hsr, const unsigned* hsc,
                         const unsigned* nb, const unsigned* bsr, const unsigned* bsc,
                         const unsigned long long* amin, unsigned* imin) {
    long long i = (long long)blockIdx.x * blockDim.x + threadIdx.x;
    if (i >= (long long)SL * IMG) return;
    int s = (int)(i / IMG);
    int r = (int)(i % IMG);
    int ry = r / WW, rx = r % WW;
    float lbl = lab[i];
    int e; double cr, cc2;
    if (!entry_centroid(lbl, s, hkey, hcnt, hsr, hsc, nb, bsr, bsc, &e, &cr, &cc2)) return;
    double ang = atan2((double)rx - cc2, (double)ry - cr);
    if (enc_d(ang) == amin[(size_t)s * ATS + e])
        atomicMin(&imin[(size_t)s * ATS + e], (unsigned)(ry * WW + rx));
}

// ============================================================
// K5: compact entries with count>=MINPTS (incl. background label 0),
//     sort ascending by label (np.unique order), emit centers + R.
//     Lists are tiny; one thread per slice.
// ============================================================
__global__ void k_compact(const unsigned* hkey, const unsigned* hcnt,
                          const unsigned* hsr, const unsigned* hsc,
                          const unsigned* nb, const unsigned* bsr, const unsigned* bsc,
                          const unsigned* imin, unsigned* listn,
                          double* lrow, double* lcol, double* lrad,
                          float* lfr, float* lfc) {
    int s = blockIdx.x;
    if (threadIdx.x != 0) return;
    float  lb[MAXC];
    double rw[MAXC], cl[MAXC], rd[MAXC];
    int n = 0;
    unsigned nbs = nb[s];
    if (nbs != 0u) {
        // background entry (label 0.0): always >= MINPTS in practice
        unsigned bc = (unsigned)IMG - nbs;
        if (bc >= MINPTS) {
            unsigned idx = imin[(size_t)s * ATS + TS];
            double r0 = 0.0, c0 = 0.0;
            if (idx != 0xFFFFFFFFu) { r0 = (double)(idx / WW); c0 = (double)(idx % WW); }
            lb[n] = 0.0f;
            rw[n] = (SUMT - (double)bsr[s]) / (double)bc;
            cl[n] = (SUMT - (double)bsc[s]) / (double)bc;
            rd[n] = sqrt(r0 * r0 + c0 * c0);
            ++n;
        }
        for (int t = 0; t < TS && n < MAXC; ++t) {
            unsigned k = hkey[(size_t)s * TS + t];
            if (k == 0u) continue;
            unsigned c = hcnt[(size_t)s * TS + t];
            if (c < MINPTS) continue;
            unsigned idx = imin[(size_t)s * ATS + t];
            double r0 = 0.0, c0 = 0.0;
            if (idx != 0xFFFFFFFFu) { r0 = (double)(idx / WW); c0 = (double)(idx % WW); }
            lb[n] = __uint_as_float(k);
            rw[n] = (double)hsr[(size_t)s * TS + t] / (double)c;
            cl[n] = (double)hsc[(size_t)s * TS + t] / (double)c;
            rd[n] = sqrt(r0 * r0 + c0 * c0);
            ++n;
        }
        // insertion sort ascending by label (labels distinct; 0.0 lands first)
        for (int a = 1; a < n; ++a) {
            float  kl = lb[a]; double kr = rw[a], kc = cl[a], kd = rd[a];
            int bpos = a - 1;
            while (bpos >= 0 && lb[bpos] > kl) {
                lb[bpos + 1] = lb[bpos]; rw[bpos + 1] = rw[bpos];
                cl[bpos + 1] = cl[bpos]; rd[bpos + 1] = rd[bpos];
                --bpos;
            }
            lb[bpos + 1] = kl; rw[bpos + 1] = kr; cl[bpos + 1] = kc; rd[bpos + 1] = kd;
        }
    }
    listn[s] = (unsigned)n;
    for (int a = 0; a < n; ++a) {   // arrays beyond n stay zero (k_reset) for WMMA padding
        lrow[(size_t)s * MAXC + a] = rw[a];
        lcol[(size_t)s * MAXC + a] = cl[a];
        lrad[(size_t)s * MAXC + a] = rd[a];
        lfr [(size_t)s * MAXC + a] = (float)rw[a];
        lfc [(size_t)s * MAXC + a] = (float)cl[a];
    }
}

// ============================================================
// M1: pairwise tgt·pred center dot products via V_WMMA_F32_16X16X4_F32.
//     A: 16x4 (K = row,col,0,0) lanes0-15 carry K0,K1; lanes16-31 zero (K2,K3).
//     B: 4x16 mirrored. D 16x16 f32: VGPR r -> M=r (lanes<16) / M=r+8.
//     One wave per batch pair; counts are block-uniform (EXEC all-1s at WMMA).
// ============================================================
__global__ __launch_bounds__(32) void k_wmma_dot(const unsigned* __restrict__ listn,
                                                 const float* __restrict__ lfr,
                                                 const float* __restrict__ lfc,
                                                 float* __restrict__ dotm) {
    int b = blockIdx.x;
    int lane = threadIdx.x;
    int nt  = (int)listn[8 + b];
    int npp = (int)listn[b];
    if (nt == 0 || npp == 0) return;
    int nti = (nt + 15) >> 4, ntj = (npp + 15) >> 4;
    int l15 = lane & 15;
    for (int ti = 0; ti < nti; ++ti) {
        for (int tj = 0; tj < ntj; ++tj) {
            int i0 = ti * 16, j0 = tj * 16;
            v2f a  = {0.f, 0.f};
            v2f bb = {0.f, 0.f};
            if (lane < 16) {
                a.x  = lfr[(size_t)(8 + b) * MAXC + i0 + l15];
                a.y  = lfc[(size_t)(8 + b) * MAXC + i0 + l15];
                bb.x = lfr[(size_t)b * MAXC + j0 + l15];
                bb.y = lfc[(size_t)b * MAXC + j0 + l15];
            }
            v8f c = {0.f, 0.f, 0.f, 0.f, 0.f, 0.f, 0.f, 0.f};
            c = __builtin_amdgcn_wmma_f32_16x16x4_f32(
                    false, a, false, bb, (short)0, c, false, false);
            int M0 = (lane < 16) ? 0 : 8;
            #pragma unroll
            for (int r2 = 0; r2 < 8; ++r2) {
                int Mi = i0 + M0 + r2;
                int Nj = j0 + l15;
                dotm[((size_t)b * MAXC + Mi) * MAXC + Nj] = c[r2];
            }
        }
    }
}

// ============================================================
// M2: nearest-centroid matching (d = |t|^2+|p|^2-2*dot from WMMA),
//     S += 3/(2pi) * |R_p - R_t| per matched pair; out = f32(S)*0.5*w^2.
// ============================================================
__global__ void k_match(const unsigned* __restrict__ listn,
                        const double* __restrict__ lrad,
                        const float* __restrict__ lfr, const float* __restrict__ lfc,
                        const float* __restrict__ dotm,
                        const float* __restrict__ wptr, float* __restrict__ out) {
    if (blockIdx.x != 0 || threadIdx.x != 0) return;
    const double CK = 1.5 / 3.14159265358979323846;   // 1/pi + 1/(2*pi)
    double S = 0.0;
    for (int b = 0; b < 8; ++b) {
        int nt  = (int)listn[8 + b];
        int npp = (int)listn[b];
        if (nt == 0 || npp == 0) continue;            // sample_stats -> None
        if (nt <= npp) {
            for (int i = 0; i < nt; ++i) {
                float tr = lfr[(size_t)(8 + b) * MAXC + i], tc = lfc[(size_t)(8 + b) * MAXC + i];
                float t2 = tr * tr + tc * tc;
                float bd = 3.4e38f; int bj = 0;
                for (int j = 0; j < npp; ++j) {
                    float pr = lfr[(size_t)b * MAXC + j], pc = lfc[(size_t)b * MAXC + j];
                    float d = t2 + pr * pr + pc * pc
                              - 2.f * dotm[((size_t)b * MAXC + i) * MAXC + j];
                    if (d < bd) { bd = d; bj = j; }   // first-occurrence min, like np.argmin
                }
                S += CK * fabs(lrad[(size_t)b * MAXC + bj] - lrad[(size_t)(8 + b) * MAXC + i]);
            }
        } else {
            for (int j = 0; j < npp; ++j) {
                float pr = lfr[(size_t)b * MAXC + j], pc = lfc[(size_t)b * MAXC + j];
                float p2 = pr * pr + pc * pc;
                float bd = 3.4e38f; int bi = 0;
                for (int i = 0; i < nt; ++i) {
                    float tr = lfr[(size_t)(8 + b) * MAXC + i], tc = lfc[(size_t)(8 + b) * MAXC + i];
                    float d = p2 + tr * tr + tc * tc
                              - 2.f * dotm[((size_t)b * MAXC + i) * MAXC + j];
                    if (d < bd) { bd = d; bi = i; }
                }
                S += CK * fabs(lrad[(size_t)(8 + b) * MAXC + bi] - lrad[(size_t)b * MAXC + j]);
            }
        }
    }
    float w = wptr[0];
    out[0] = (float)S * (0.5f * (w * w));
}

// K0: zero/initialize all accumulation state every call (determinism).
__global__ void k_reset(unsigned* hkey, unsigned* hcnt, unsigned* hsr, unsigned* hsc,
                        unsigned long long* amin, unsigned* imin,
                        unsigned* nb, unsigned* bsr, unsigned* bsc, unsigned* listn,
                        double* lrow, double* lcol, double* lrad, float* lfr, float* lfc) {
    int t = blockIdx.x * blockDim.x + threadIdx.x;
    if (t < SL * TS)   { hkey[t] = 0u; hcnt[t] = 0u; hsr[t] = 0u; hsc[t] = 0u; }
    if (t < SL * ATS)  { amin[t] = ~0ULL; imin[t] = 0xFFFFFFFFu; }
    if (t < SL)        { nb[t] = 0u; bsr[t] = 0u; bsc[t] = 0u; listn[t] = 0u; }
    if (t < SL * MAXC) { lrow[t] = 0.0; lcol[t] = 0.0; lrad[t] = 0.0; lfr[t] = 0.f; lfc[t] = 0.f; }
}

// ============================================================
// host-side orchestration (graph-capture safe: kernels only)
// workspace requirement ~44 MB
// ============================================================
extern "C" void kernel_launch(void* const* d_in, const int* in_sizes, int n_in,
                              void* d_out, int out_size, void* d_ws, size_t ws_size,
                              hipStream_t stream) {
    const float* pred = (const float*)d_in[0];   // [8,1,512,512] logits
    const float* tgt  = (const float*)d_in[1];   // [8,1,512,512] binary
    const float* wgt  = (const float*)d_in[2];   // scalar

    char* ws = (char*)d_ws;
    size_t off = 0;
    auto take = [&](size_t bytes) { size_t r = off; off = (off + bytes + 255) & ~(size_t)255; return r; };

    float*              W0v  = (float*)             (ws + take((size_t)SL * SLC * 4));
    float*              W1v  = (float*)             (ws + take((size_t)SL * SLC * 4));
    unsigned*           HKEY = (unsigned*)          (ws + take((size_t)SL * TS * 4));
    unsigned*           HCNT = (unsigned*)          (ws + take((size_t)SL * TS * 4));
    unsigned*           HSR  = (unsigned*)          (ws + take((size_t)SL * TS * 4));
    unsigned*           HSC  = (unsigned*)          (ws + take((size_t)SL * TS * 4));
    unsigned long long* AMIN = (unsigned long long*)(ws + take((size_t)SL * ATS * 8));
    unsigned*           IMIN = (unsigned*)          (ws + take((size_t)SL * ATS * 4));
    unsigned*           NB   = (unsigned*)          (ws + take((size_t)SL * 4));
    unsigned*           BSR  = (unsigned*)          (ws + take((size_t)SL * 4));
    unsigned*           BSC  = (unsigned*)          (ws + take((size_t)SL * 4));
    unsigned*           LSTN = (unsigned*)          (ws + take((size_t)SL * 4));
    double*             LROW = (double*)            (ws + take((size_t)SL * MAXC * 8));
    double*             LCOL = (double*)            (ws + take((size_t)SL * MAXC * 8));
    double*             LRAD = (double*)            (ws + take((size_t)SL * MAXC * 8));
    float*              LFR  = (float*)             (ws + take((size_t)SL * MAXC * 4));
    float*              LFC  = (float*)             (ws + take((size_t)SL * MAXC * 4));
    float*              DOT  = (float*)             (ws + take((size_t)BATCH * MAXC * MAXC * 4));
    (void)ws_size; (void)in_sizes; (void)n_in; (void)out_size;

    // seed + reset state
    {
        long long tot = (long long)SL * SLC;
        k_init<<<(unsigned)((tot + 255) / 256), 256, 0, stream>>>(pred, tgt, W0v, W1v);
    }
    {
        int tot = SL * ATS;
        k_reset<<<(tot + 255) / 256, 256, 0, stream>>>(HKEY, HCNT, HSR, HSC, AMIN, IMIN,
                                                       NB, BSR, BSC, LSTN,
                                                       LROW, LCOL, LRAD, LFR, LFC);
    }

    // 500 masked max-pool iterations, 16 per launch (halo-16 temporal blocking)
    float* cur = W0v; float* nxt = W1v;
    int done = 0;
    while (done < 500) {
        int it = (500 - done < HALO) ? (500 - done) : HALO;
        k_prop<<<dim3(NTIL, NTIL, SL), dim3(16, 16), 0, stream>>>(cur, nxt, it);
        float* t = cur; cur = nxt; nxt = t;
        done += it;
    }

    // contour labels (reuse the stale ping-pong buffer for the label image)
    float* lab = nxt;
    {
        long long tot = (long long)SL * IMG;
        unsigned g = (unsigned)((tot + 255) / 256);
        k_contour<<<g, 256, 0, stream>>>(cur, lab, HKEY, HCNT, HSR, HSC, NB, BSR, BSC);
        k_angmin <<<g, 256, 0, stream>>>(lab, HKEY, HCNT, HSR, HSC, NB, BSR, BSC, AMIN);
        k_idxmin <<<g, 256, 0, stream>>>(lab, HKEY, HCNT, HSR, HSC, NB, BSR, BSC, AMIN, IMIN);
    }
    k_compact<<<SL, 1, 0, stream>>>(HKEY, HCNT, HSR, HSC, NB, BSR, BSC, IMIN, LSTN,
                                    LROW, LCOL, LRAD, LFR, LFC);
    k_wmma_dot<<<BATCH, 32, 0, stream>>>(LSTN, LFR, LFC, DOT);
    k_match<<<1, 1, 0, stream>>>(LSTN, LRAD, LFR, LFC, DOT, wgt, (float*)d_out);
}